// FieldAwareFactorizationMachine_21912923144762
// MI455X (gfx1250) — compile-verified
//
#include <hip/hip_runtime.h>
#include <math.h>

typedef float v2f __attribute__((ext_vector_type(2)));
typedef float v8f __attribute__((ext_vector_type(8)));

#define NUM_FIELDS 24
#define FIELD_DIM 50000
#define TOTAL_VOCAB (NUM_FIELDS * FIELD_DIM)
#define EMB 8
#define BATCH 4096
#define NPAIRS 276                       // 24*23/2 upper-triangular pairs
#define SAMPLES_PER_BLOCK 16
#define THREADS 256
#define WAVES (THREADS / 32)
#define TABLE_BYTES (TOTAL_VOCAB * EMB * 4)   // 38,400,000 B per field table
#define ROW_BYTES (EMB * 4)                   // 32 B per embedding row

__global__ __launch_bounds__(THREADS)
void ffm_wmma_kernel(const int* __restrict__ x,
                     const float* __restrict__ Wc,
                     const float* __restrict__ Wl,
                     const float* __restrict__ bias,
                     float* __restrict__ out)
{
    __shared__ int   s_idxb[SAMPLES_PER_BLOCK][NUM_FIELDS];  // byte offset of row
    __shared__ int   s_pair[NPAIRS];                         // i | (j<<8)
    __shared__ float s_lin[SAMPLES_PER_BLOCK];
    __shared__ float s_cross[SAMPLES_PER_BLOCK];

    const int tid  = threadIdx.x;
    const int wave = tid >> 5;
    const int lane = tid & 31;
    const int s    = lane & 15;   // sample slot within 16-sample tile
    const int h    = lane >> 4;   // K half (lanes 0-15: K{0,1}; 16-31: K{2,3})
    const int sample0 = blockIdx.x * SAMPLES_PER_BLOCK;
    const char* WcB = (const char*)Wc;

    if (tid < SAMPLES_PER_BLOCK) { s_lin[tid] = 0.f; s_cross[tid] = 0.f; }

    // Stage vocab indices as byte offsets within one field table (idx*32 < 2^26).
    for (int e = tid; e < SAMPLES_PER_BLOCK * NUM_FIELDS; e += THREADS) {
        int ss = e / NUM_FIELDS, f = e % NUM_FIELDS;
        s_idxb[ss][f] = (x[(size_t)(sample0 + ss) * NUM_FIELDS + f]
                         + f * FIELD_DIM) * ROW_BYTES;
    }
    // Decode each flat pair index -> (i, j) once; stash in LDS.
    for (int p = tid; p < NPAIRS; p += THREADS) {
        int i = 0, rem = p;
        while (rem >= (NUM_FIELDS - 1 - i)) { rem -= (NUM_FIELDS - 1 - i); ++i; }
        int j = i + 1 + rem;
        s_pair[p] = i | (j << 8);
    }
    __syncthreads();

    // Linear term: 24 gathered scalars per sample, LDS float atomics.
    for (int e = tid; e < SAMPLES_PER_BLOCK * NUM_FIELDS; e += THREADS) {
        int ss = e / NUM_FIELDS, f = e % NUM_FIELDS;
        atomicAdd(&s_lin[ss], Wl[s_idxb[ss][f] >> 5]);  // /ROW_BYTES -> element idx
    }

    // Cross term: per pair (i<j), D = A x B + C with
    //   A[m, k] = Wc[j, idx[sample m, i], k]   (16x8, two K=4 steps)
    //   B[k, n] = Wc[i, idx[sample n, j], k]
    // diag(C) accumulates the per-sample FFM pair dot products.
    v8f acc = {};

#pragma unroll 2
    for (int p = wave; p < NPAIRS; p += WAVES) {
        int pk = s_pair[p];
        int i  = pk & 255;
        int j  = pk >> 8;

        // 32-bit byte offsets from the uniform W_cross base (total table < 2^31).
        int offA = j * TABLE_BYTES + s_idxb[s][i] + 8 * h;
        int offB = i * TABLE_BYTES + s_idxb[s][j] + 8 * h;

        // This lane supplies K = {2h, 2h+1} (step 0) and K = {4+2h, 5+2h} (step 1).
        v2f a0 = *(const v2f*)(WcB + offA);
        v2f a1 = *(const v2f*)(WcB + offA + 16);
        v2f b0 = *(const v2f*)(WcB + offB);
        v2f b1 = *(const v2f*)(WcB + offB + 16);

        acc = __builtin_amdgcn_wmma_f32_16x16x4_f32(false, a0, false, b0,
                                                    (short)0, acc, false, false);
        acc = __builtin_amdgcn_wmma_f32_16x16x4_f32(false, a1, false, b1,
                                                    (short)0, acc, false, false);
    }

    // Extract diagonal of the 16x16 f32 accumulator:
    //   C[m,n]: vgpr r holds M = r + 8*(lane>=16), N = lane%16.
    //   diag lives at (r, lane=r) for r<8 and (r, lane=r+24) for r>=8.
    float diag = 0.f;
    {
        const int n    = lane & 15;
        const int mofs = (lane >= 16) ? 8 : 0;
#pragma unroll
        for (int r = 0; r < 8; ++r)
            if (r + mofs == n) diag = acc[r];
    }
    atomicAdd(&s_cross[s], diag);   // non-diag lanes add 0.f
    __syncthreads();

    if (tid < SAMPLES_PER_BLOCK) {
        float z = s_lin[tid] + s_cross[tid] + bias[0];
        out[sample0 + tid] = 1.0f / (1.0f + expf(-z));
    }
}

extern "C" void kernel_launch(void* const* d_in, const int* in_sizes, int n_in,
                              void* d_out, int out_size, void* d_ws, size_t ws_size,
                              hipStream_t stream)
{
    const int*   x    = (const int*)d_in[0];
    const float* Wc   = (const float*)d_in[1];
    const float* Wl   = (const float*)d_in[2];
    const float* bias = (const float*)d_in[3];
    float*       out  = (float*)d_out;

    dim3 grid(BATCH / SAMPLES_PER_BLOCK);   // 256 blocks
    dim3 block(THREADS);                    // 8 wave32 per block
    ffm_wmma_kernel<<<grid, block, 0, stream>>>(x, Wc, Wl, bias, out);
}